// SubconsciousCore_64690797412875
// MI455X (gfx1250) — compile-verified
//
#include <hip/hip_runtime.h>
#include <math.h>

typedef __attribute__((ext_vector_type(2))) float v2f;
typedef __attribute__((ext_vector_type(4))) float v4f;
typedef __attribute__((ext_vector_type(8))) float v8f;

#define D 2048
#define NROWS 65536
#define KMEM 8
#define MROWS 12
#define MPAD 16
#define EPS 1e-12f

// ---- workspace layout (float element offsets) ----
#define OFF_SCORES 0          // 65536 floats ; reused as PC after top-k
#define OFF_PC     0          // 8 splits * 16 m * 2048 = 262144 floats
#define OFF_IDX    262144     // 16 ints
#define OFF_C      262160     // 16 * 2048
#define OFF_QVEC   294928     // 2048
#define OFF_TVEC   296976     // 2048
#define OFF_SRAW   299024     // 2048
#define OFF_G      301072     // 1 (+pad)
#define OFF_ATTN   301088     // 16
#define OFF_PQ     301104     // 16 * 2048
#define OFF_PG     333872     // 16 * 2048
#define OFF_PM     301104     // reuse PQ region (8 * 2048 used)

// ---------------- kernel 1: cosine ranking scores ----------------
// one wave per row; score = (row . z) / max(||row||, eps)  (constant 1/||z|| dropped: monotone).
// mem_bank is a 512MB single-use stream > 192MB L2: load non-temporal so the
// 96MB of weight matrices stay L2-resident for the later stages / replays.
__global__ void k_scores(const float* __restrict__ mem, const float* __restrict__ z,
                         float* __restrict__ scores) {
    int wave = (blockIdx.x * blockDim.x + threadIdx.x) >> 5;
    int lane = threadIdx.x & 31;
    const v4f* row = (const v4f*)(mem + (size_t)wave * D);
    const v4f* zv  = (const v4f*)z;
    float s = 0.f, n = 0.f;
    #pragma unroll 4
    for (int j = lane; j < D / 4; j += 32) {
        v4f r = __builtin_nontemporal_load(&row[j]);   // global_load_b128 th:NT
        v4f q = zv[j];                                  // z is hot, keep RT
        s += r.x * q.x + r.y * q.y + r.z * q.z + r.w * q.w;
        n += r.x * r.x + r.y * r.y + r.z * r.z + r.w * r.w;
    }
    for (int o = 16; o > 0; o >>= 1) {
        s += __shfl_xor(s, o, 32);
        n += __shfl_xor(n, o, 32);
    }
    if (lane == 0) {
        float sc = s / fmaxf(sqrtf(n), EPS);
        __builtin_nontemporal_store(sc, &scores[wave]);
    }
}

// ---------------- kernel 2: deterministic top-8 (argmax x8, ties -> lower idx) ----------------
__global__ void k_topk(const float* __restrict__ scores, int* __restrict__ idx) {
    __shared__ float sv[1024];
    __shared__ int   si[1024];
    __shared__ int   chosen[KMEM];
    int tid = threadIdx.x;
    for (int k = 0; k < KMEM; ++k) {
        float best = -3.4e38f; int bi = NROWS;
        for (int i = tid; i < NROWS; i += 1024) {
            bool skip = false;
            for (int c = 0; c < k; ++c) if (chosen[c] == i) skip = true;
            if (skip) continue;
            float v = scores[i];
            if (v > best || (v == best && i < bi)) { best = v; bi = i; }
        }
        sv[tid] = best; si[tid] = bi; __syncthreads();
        for (int s = 512; s > 0; s >>= 1) {
            if (tid < s) {
                float ov = sv[tid + s]; int oi = si[tid + s];
                if (ov > sv[tid] || (ov == sv[tid] && oi < si[tid])) { sv[tid] = ov; si[tid] = oi; }
            }
            __syncthreads();
        }
        if (tid == 0) { chosen[k] = si[0]; idx[k] = si[0]; }
        __syncthreads();
    }
}

// ---------------- kernel 3: build C (protos raw | normalized dreams | zero pad) ----------------
__global__ void k_buildC(const float* __restrict__ mem, const float* __restrict__ z,
                         const float* __restrict__ noise, const int* __restrict__ idx,
                         float* __restrict__ C) {
    int r = blockIdx.x, tid = threadIdx.x;
    float* dst = C + (size_t)r * D;
    if (r < KMEM) {
        const float* src = mem + (size_t)idx[r] * D;
        for (int j = tid; j < D; j += 256) dst[j] = src[j];
    } else if (r < MROWS) {
        const float* nz = noise + (size_t)(r - KMEM) * D;
        __shared__ float red[256];
        float ss = 0.f;
        for (int j = tid; j < D; j += 256) {
            float v = 0.7f * nz[j] + 0.3f * z[j];
            ss += v * v;
        }
        red[tid] = ss; __syncthreads();
        for (int s = 128; s > 0; s >>= 1) { if (tid < s) red[tid] += red[tid + s]; __syncthreads(); }
        float inv = 1.0f / fmaxf(sqrtf(red[0]), EPS);
        for (int j = tid; j < D; j += 256) {
            float v = 0.7f * nz[j] + 0.3f * z[j];
            dst[j] = v * inv;
        }
    } else {
        for (int j = tid; j < D; j += 256) dst[j] = 0.0f;
    }
}

// ---------------- kernel 4: split-K partials for ctx@Wq and ctx@Wg1 ----------------
__global__ void k_ctx_partial(const float* __restrict__ z, const float* __restrict__ h,
                              const float* __restrict__ Wq, const float* __restrict__ Wg1,
                              float* __restrict__ PQ, float* __restrict__ PG) {
    int tid = threadIdx.x;
    int j   = blockIdx.x * 256 + tid;
    int i0  = blockIdx.y * 256;
    const float* W = blockIdx.z ? Wg1 : Wq;
    float*       P = blockIdx.z ? PG  : PQ;
    __shared__ float sctx[256];
    int ii = i0 + tid;
    sctx[tid] = (ii < D) ? z[ii] : h[ii - D];
    __syncthreads();
    float acc = 0.f;
    const float* col = W + (size_t)i0 * D + j;
    #pragma unroll 8
    for (int t = 0; t < 256; ++t) acc += sctx[t] * col[(size_t)t * D];
    P[(size_t)blockIdx.y * D + j] = acc;
}

// ---------------- kernel 5: reduce ctx partials + tanh ----------------
__global__ void k_ctx_reduce(const float* __restrict__ PQ, const float* __restrict__ PG,
                             const float* __restrict__ bq, const float* __restrict__ bg1,
                             float* __restrict__ qvec, float* __restrict__ tvec) {
    int j = blockIdx.x * 256 + threadIdx.x;
    float q = bq[j], t = bg1[j];
    #pragma unroll
    for (int s = 0; s < 16; ++s) { q += PQ[(size_t)s * D + j]; t += PG[(size_t)s * D + j]; }
    qvec[j] = tanhf(q);
    tvec[j] = tanhf(t);
}

// ---------------- kernel 6: gate scalar ----------------
__global__ void k_gate(const float* __restrict__ tvec, const float* __restrict__ wg2,
                       const float* __restrict__ bg2, float* __restrict__ gout) {
    __shared__ float red[256];
    int tid = threadIdx.x;
    float a = 0.f;
    for (int j = tid; j < D; j += 256) a += tvec[j] * wg2[j];
    red[tid] = a; __syncthreads();
    for (int s = 128; s > 0; s >>= 1) { if (tid < s) red[tid] += red[tid + s]; __syncthreads(); }
    if (tid == 0) gout[0] = 1.0f / (1.0f + expf(-(red[0] + bg2[0])));
}

// ---------------- kernel 7: WMMA f32 16x16x4 GEMM, C(16x2048) @ Wc(2048x2048), K split 8 ----------------
// wave w -> column tile (w>>3)*16, K-chunk (w&7)*256. A = C tile, B = Wc tile (ISA operand layouts).
__global__ void k_wc_wmma(const float* __restrict__ Cmat, const float* __restrict__ Wc,
                          float* __restrict__ PC) {
    int tid   = threadIdx.x;
    int w     = blockIdx.x * (blockDim.x >> 5) + (tid >> 5);
    int tile  = w >> 3;
    int split = w & 7;
    int lane  = tid & 31;
    int half  = lane >> 4;       // 0: K pair {0,1}, 1: K pair {2,3}
    int lm    = lane & 15;       // A row (M) / B column (N)
    int n0    = tile * 16;
    const float* Crow = Cmat + (size_t)lm * D;
    int kk0 = split * 256;
    v8f acc = {0.f, 0.f, 0.f, 0.f, 0.f, 0.f, 0.f, 0.f};
    for (int kk = kk0; kk < kk0 + 256; kk += 4) {
        int ka = kk + (half ? 2 : 0);
        v2f a, b;
        a.x = Crow[ka];
        a.y = Crow[ka + 1];
        b.x = Wc[(size_t)ka * D + n0 + lm];
        b.y = Wc[(size_t)(ka + 1) * D + n0 + lm];
        acc = __builtin_amdgcn_wmma_f32_16x16x4_f32(false, a, false, b, (short)0, acc,
                                                    false, false);
    }
    #pragma unroll
    for (int v = 0; v < 8; ++v) {
        int m = v + (half ? 8 : 0);          // D layout: VGPR v = row v (lanes 0-15) / v+8 (16-31)
        PC[(size_t)(split * MPAD + m) * D + n0 + lm] = acc[v];
    }
}

// ---------------- kernel 8: logits + softmax -> attn ----------------
__global__ void k_logits(const float* __restrict__ PC, const float* __restrict__ bc,
                         const float* __restrict__ qvec, const float* __restrict__ wsv,
                         const float* __restrict__ bs, float* __restrict__ attn) {
    __shared__ float red[256];
    __shared__ float fin[MROWS];
    int tid = threadIdx.x;
    float accm[MROWS];
    #pragma unroll
    for (int m = 0; m < MROWS; ++m) accm[m] = 0.f;
    for (int n = tid; n < D; n += 256) {
        float u = qvec[n] * wsv[n];
        #pragma unroll
        for (int m = 0; m < MROWS; ++m) {
            float cp = bc[n];
            #pragma unroll
            for (int s = 0; s < 8; ++s) cp += PC[(size_t)(s * MPAD + m) * D + n];
            accm[m] += cp * u;
        }
    }
    for (int m = 0; m < MROWS; ++m) {
        red[tid] = accm[m]; __syncthreads();
        for (int s = 128; s > 0; s >>= 1) { if (tid < s) red[tid] += red[tid + s]; __syncthreads(); }
        if (tid == 0) fin[m] = red[0];
        __syncthreads();
    }
    if (tid == 0) {
        float lg[MROWS], mx = -3.4e38f;
        for (int m = 0; m < MROWS; ++m) { lg[m] = fin[m] + bs[0]; mx = fmaxf(mx, lg[m]); }
        float sum = 0.f;
        for (int m = 0; m < MROWS; ++m) { lg[m] = expf(lg[m] - mx); sum += lg[m]; }
        for (int m = 0; m < MROWS; ++m) attn[m] = lg[m] / sum;
    }
}

// ---------------- kernel 9: s_raw = attn @ C ----------------
__global__ void k_sraw(const float* __restrict__ attn, const float* __restrict__ Cmat,
                       float* __restrict__ sraw) {
    int d = blockIdx.x * 256 + threadIdx.x;
    float s = 0.f;
    #pragma unroll
    for (int m = 0; m < MROWS; ++m) s += attn[m] * Cmat[(size_t)m * D + d];
    sraw[d] = s;
}

// ---------------- kernel 10: split-K partials for s_raw @ Wm ----------------
__global__ void k_wm_partial(const float* __restrict__ sraw, const float* __restrict__ Wm,
                             float* __restrict__ PM) {
    int tid = threadIdx.x;
    int j   = blockIdx.x * 256 + tid;
    int i0  = blockIdx.y * 256;
    __shared__ float sx[256];
    sx[tid] = sraw[i0 + tid];
    __syncthreads();
    float acc = 0.f;
    const float* col = Wm + (size_t)i0 * D + j;
    #pragma unroll 8
    for (int t = 0; t < 256; ++t) acc += sx[t] * col[(size_t)t * D];
    PM[(size_t)blockIdx.y * D + j] = acc;
}

// ---------------- kernel 11: out = g * tanh(s_raw@Wm + bm) ----------------
__global__ void k_out(const float* __restrict__ PM, const float* __restrict__ bm,
                      const float* __restrict__ g, float* __restrict__ out) {
    int j = blockIdx.x * 256 + threadIdx.x;
    float a = bm[j];
    #pragma unroll
    for (int s = 0; s < 8; ++s) a += PM[(size_t)s * D + j];
    out[j] = g[0] * tanhf(a);
}

extern "C" void kernel_launch(void* const* d_in, const int* in_sizes, int n_in,
                              void* d_out, int out_size, void* d_ws, size_t ws_size,
                              hipStream_t stream) {
    const float* z_t   = (const float*)d_in[0];
    const float* h_t   = (const float*)d_in[1];
    const float* mem   = (const float*)d_in[2];
    const float* noise = (const float*)d_in[3];
    const float* Wq    = (const float*)d_in[4];
    const float* bq    = (const float*)d_in[5];
    const float* Wc    = (const float*)d_in[6];
    const float* bc    = (const float*)d_in[7];
    const float* wsv   = (const float*)d_in[8];
    const float* bs    = (const float*)d_in[9];
    const float* Wm    = (const float*)d_in[10];
    const float* bm    = (const float*)d_in[11];
    const float* Wg1   = (const float*)d_in[12];
    const float* bg1   = (const float*)d_in[13];
    const float* wg2   = (const float*)d_in[14];
    const float* bg2   = (const float*)d_in[15];
    float* out  = (float*)d_out;
    float* base = (float*)d_ws;

    float* scores = base + OFF_SCORES;
    int*   idx    = (int*)(base + OFF_IDX);
    float* Cmat   = base + OFF_C;
    float* qvec   = base + OFF_QVEC;
    float* tvec   = base + OFF_TVEC;
    float* sraw   = base + OFF_SRAW;
    float* gscal  = base + OFF_G;
    float* attn   = base + OFF_ATTN;
    float* PQ     = base + OFF_PQ;
    float* PG     = base + OFF_PG;
    float* PC     = base + OFF_PC;
    float* PM     = base + OFF_PM;

    // 1) 512MB bandwidth sweep: cosine ranking scores (8 waves/block, 1 wave/row, NT loads)
    k_scores<<<NROWS / 8, 256, 0, stream>>>(mem, z_t, scores);
    // 2) top-8 indices
    k_topk<<<1, 1024, 0, stream>>>(scores, idx);
    // 3) candidate matrix C (16 x 2048, rows 12..15 zero)
    k_buildC<<<MPAD, 256, 0, stream>>>(mem, z_t, noise, idx, Cmat);
    // 4) ctx GEMV partials for Wq / Wg1 (K=4096 split 16)
    k_ctx_partial<<<dim3(8, 16, 2), 256, 0, stream>>>(z_t, h_t, Wq, Wg1, PQ, PG);
    // 5) reduce + tanh -> q_vec, t_vec
    k_ctx_reduce<<<8, 256, 0, stream>>>(PQ, PG, bq, bg1, qvec, tvec);
    // 6) gate scalar
    k_gate<<<1, 256, 0, stream>>>(tvec, wg2, bg2, gscal);
    // 7) WMMA GEMM: 128 column tiles x 8 K-splits = 1024 waves (8 waves/block)
    k_wc_wmma<<<128, 256, 0, stream>>>(Cmat, Wc, PC);
    // 8) logits + softmax
    k_logits<<<1, 256, 0, stream>>>(PC, bc, qvec, wsv, bs, attn);
    // 9) s_raw
    k_sraw<<<8, 256, 0, stream>>>(attn, Cmat, sraw);
    // 10) s_raw @ Wm partials (K=2048 split 8)
    k_wm_partial<<<dim3(8, 8), 256, 0, stream>>>(sraw, Wm, PM);
    // 11) final output
    k_out<<<8, 256, 0, stream>>>(PM, bm, gscal, out);
}